// InstanceAwarePointMatching_35064113005040
// MI455X (gfx1250) — compile-verified
//
#include <hip/hip_runtime.h>
#include <stdint.h>

// ---------------------------------------------------------------------------
// InstanceAwarePointMatching: per-p (512) 256x256 tiles.
//   score = exp(x); top-3 per row & per column -> scatter; combine.
// Equivalent formulation: per-row / per-col 3rd-largest RAW threshold,
// then fused elementwise pass (exp is monotone, exp(x) > 0 always).
// Bandwidth-bound: 128MB in + 256MB out => ~16.5us floor @ 23.3 TB/s.
// CDNA5 path: whole 256KB tile staged in LDS (320KB/WGP) with ONE
// TDM tensor_load_to_lds per workgroup; TDM pad feature gives a 260-float
// row pitch => aligned b128 LDS reads + conflict-free column scans.
// Outputs written as non-temporal b128 stores (256MB streamed, > L2).
// ---------------------------------------------------------------------------

typedef uint32_t u32x4 __attribute__((ext_vector_type(4)));
typedef int      i32x8 __attribute__((ext_vector_type(8)));
typedef int      i32x4 __attribute__((ext_vector_type(4)));
typedef float    f32x4 __attribute__((ext_vector_type(4)));

#define R_DIM     256
#define S_DIM     256
#define ROW_PITCH 260      // floats; TDM pads +4 DWORDs per 256-DWORD row (16B-aligned rows)
#define NTHREADS  256

struct Smem {
    float tile[R_DIM * ROW_PITCH];   // padded 256x256 f32 tile
    float row_thr[R_DIM];            // 3rd-largest raw score per row
    float col_thr[S_DIM];            // 3rd-largest raw score per column
};

__device__ __forceinline__ void top3_update(float x, float& m0, float& m1, float& m2) {
    if (x > m2) {
        if (x > m0)      { m2 = m1; m1 = m0; m0 = x; }
        else if (x > m1) { m2 = m1; m1 = x; }
        else             { m2 = x; }
    }
}

__global__ __launch_bounds__(NTHREADS) void iapm_topk_fused_kernel(
    const float*         __restrict__ score,     // [P,R,S] raw log-scores
    const unsigned char* __restrict__ refm,      // [P,R] bool
    const unsigned char* __restrict__ srcm,      // [P,S] bool
    float*               __restrict__ out_score, // [P,R,S]
    float*               __restrict__ out_corr)  // [P,R,S] (bool as 0/1 f32)
{
    __shared__ Smem smem;
    const int p = blockIdx.x;
    const int t = threadIdx.x;

    // ---- Stage tile into LDS via Tensor Data Mover (one op per workgroup).
    // D# packed per CDNA5 ISA.
    if (t < 32) {  // wave 0 only issues the TDM op (EXEC is ignored by TENSOR ops)
        uint64_t ga      = (uint64_t)(uintptr_t)(score + (size_t)p * R_DIM * S_DIM);
        uint32_t lds_off = (uint32_t)(uintptr_t)(&smem.tile[0]); // low 32b of flat addr = LDS offset

        u32x4 g0;
        g0[0] = 1u;                                   // count=1 (valid descriptor), user mode
        g0[1] = lds_off;                              // lds_addr (bytes)
        g0[2] = (uint32_t)(ga & 0xFFFFFFFFu);         // global_addr[31:0]
        g0[3] = (uint32_t)((ga >> 32) & 0x1FFFFFFu)   // global_addr[56:32]
              | (2u << 30);                           // type = 2 ("image")

        i32x8 g1;
        g1[0] = (2 << 16)                             // data_size = 4 bytes
              | (1 << 20)                             // pad_enable
              | (7 << 22)                             // pad_interval = 256 DWORDs
              | (3 << 25);                            // pad_amount  = 4 DWORDs -> pitch 260
        g1[1] = (S_DIM & 0xFFFF) << 16;               // tensor_dim0[15:0] = 256
        g1[2] = (R_DIM & 0xFFFF) << 16;               // tensor_dim1[15:0] = 256
        g1[3] = (S_DIM & 0xFFFF) << 16;               // tile_dim0 = 256
        g1[4] = R_DIM;                                // tile_dim1 = 256, tile_dim2 = 0
        g1[5] = S_DIM;                                // tensor_dim0_stride[31:0] = 256
        g1[6] = ((R_DIM * S_DIM) & 0xFFFF) << 16;     // tensor_dim1_stride[15:0] (unused, 2D tile)
        g1[7] = (R_DIM * S_DIM) >> 16;                // tensor_dim1_stride[47:16]

        i32x4 gz4 = {0, 0, 0, 0};                     // groups 2/3 unused (<=3D tile)
        i32x8 gz8 = {0, 0, 0, 0, 0, 0, 0, 0};         // extra group (clang-23 6-arg form)
        __builtin_amdgcn_tensor_load_to_lds(g0, g1, gz4, gz4, gz8, /*cpol=*/0);
        __builtin_amdgcn_s_wait_tensorcnt(0);
    }
    __syncthreads();

    const float NEG_INF = -3.402823466e38f;

    // ---- Phase 1: row thresholds. Thread t scans row t.
    {
        float m0 = NEG_INF, m1 = NEG_INF, m2 = NEG_INF;
        const float* row = &smem.tile[t * ROW_PITCH];
        #pragma unroll 4
        for (int j = 0; j < S_DIM; ++j) top3_update(row[j], m0, m1, m2);
        smem.row_thr[t] = m2;
    }

    // ---- Phase 2: column thresholds. Thread t scans column t (stride-1
    // across lanes within a row => conflict-free).
    {
        float m0 = NEG_INF, m1 = NEG_INF, m2 = NEG_INF;
        #pragma unroll 4
        for (int r = 0; r < R_DIM; ++r) top3_update(smem.tile[r * ROW_PITCH + t], m0, m1, m2);
        smem.col_thr[t] = m2;
    }
    __syncthreads();

    // ---- Phase 3: fused elementwise output, vectorized x4 along S.
    // Thread t owns columns s0..s0+3 of row (4k + t/64); lanes of a wave are
    // contiguous within one row => 512B-per-wave b128 non-temporal stores.
    const int    s0   = (t & 63) * 4;
    const size_t base = (size_t)p * R_DIM * S_DIM;
    const size_t pR   = (size_t)p * R_DIM;

    const f32x4    cthr4 = *(const f32x4*)&smem.col_thr[s0];
    const uint32_t ms_u  = *(const uint32_t*)&srcm[(size_t)p * S_DIM + s0];
    const bool msx = (ms_u & 0x000000FFu) != 0;
    const bool msy = (ms_u & 0x0000FF00u) != 0;
    const bool msz = (ms_u & 0x00FF0000u) != 0;
    const bool msw = (ms_u & 0xFF000000u) != 0;

    #pragma unroll 4
    for (int k = 0; k < R_DIM / 4; ++k) {
        const int   r    = (k << 2) + (t >> 6);
        const float rthr = smem.row_thr[r];
        const bool  mr   = refm[pR + r] != 0;          // uniform per wave: scalar load
        const f32x4 x4   = *(const f32x4*)&smem.tile[r * ROW_PITCH + s0];

        f32x4 sc, cr;
        {   // element 0
            float e = __expf(x4.x); bool ir = x4.x >= rthr, ic = x4.x >= cthr4.x;
            sc.x = 0.5f * ((ir ? e : 0.0f) + (ic ? e : 0.0f));
            cr.x = ((ir || ic) && mr && msx) ? 1.0f : 0.0f;
        }
        {   // element 1
            float e = __expf(x4.y); bool ir = x4.y >= rthr, ic = x4.y >= cthr4.y;
            sc.y = 0.5f * ((ir ? e : 0.0f) + (ic ? e : 0.0f));
            cr.y = ((ir || ic) && mr && msy) ? 1.0f : 0.0f;
        }
        {   // element 2
            float e = __expf(x4.z); bool ir = x4.z >= rthr, ic = x4.z >= cthr4.z;
            sc.z = 0.5f * ((ir ? e : 0.0f) + (ic ? e : 0.0f));
            cr.z = ((ir || ic) && mr && msz) ? 1.0f : 0.0f;
        }
        {   // element 3
            float e = __expf(x4.w); bool ir = x4.w >= rthr, ic = x4.w >= cthr4.w;
            sc.w = 0.5f * ((ir ? e : 0.0f) + (ic ? e : 0.0f));
            cr.w = ((ir || ic) && mr && msw) ? 1.0f : 0.0f;
        }

        const size_t idx = base + (size_t)r * S_DIM + s0;    // 16B aligned
        __builtin_nontemporal_store(sc, (f32x4*)&out_score[idx]);
        __builtin_nontemporal_store(cr, (f32x4*)&out_corr[idx]);
    }
}

extern "C" void kernel_launch(void* const* d_in, const int* in_sizes, int n_in,
                              void* d_out, int out_size, void* d_ws, size_t ws_size,
                              hipStream_t stream) {
    const float*         score = (const float*)d_in[0];          // [P,R,S]
    // d_in[1] = node_corr_scores: unused (conditional=False in reference)
    const unsigned char* refm  = (const unsigned char*)d_in[2];  // [P,R] bool
    const unsigned char* srcm  = (const unsigned char*)d_in[3];  // [P,S] bool

    const int P = in_sizes[1];                                   // node_corr_scores has P elements
    float* out_score = (float*)d_out;                            // first output, P*R*S floats
    float* out_corr  = out_score + (size_t)P * R_DIM * S_DIM;    // second output, concatenated

    iapm_topk_fused_kernel<<<P, NTHREADS, 0, stream>>>(score, refm, srcm, out_score, out_corr);
}